// WassersteinLoss_76355928588616
// MI455X (gfx1250) — compile-verified
//
#include <hip/hip_runtime.h>
#include <hip/hip_bf16.h>
#include <stdint.h>

// Problem shape (from reference): B x N with N = 4096.
#define ROW_N    4096
#define THREADS  512
#define PAIRS_PER_THREAD (ROW_N / 2 / THREADS)   // 4
#define ELEMS_PER_THREAD (ROW_N / THREADS)       // 8

// Builtin signature (probe-confirmed): param0 = AS1 (global) int4*, param1 =
// AS3 (LDS) int4*, then imm offset, imm cpol.
typedef int v4i __attribute__((ext_vector_type(4)));
typedef __attribute__((address_space(1))) v4i* gvec_ptr;
typedef __attribute__((address_space(3))) v4i* lvec_ptr;

// ---- CDNA5 async global->LDS copy of one row (16 KB) ----
__device__ __forceinline__ void async_copy_row(const float* __restrict__ g,
                                               float* l, int tid) {
#if __has_builtin(__builtin_amdgcn_global_load_async_to_lds_b128)
  const int e = tid * ELEMS_PER_THREAD;
  __builtin_amdgcn_global_load_async_to_lds_b128(
      (gvec_ptr)(g + e),     (lvec_ptr)(l + e),     0, 0);
  __builtin_amdgcn_global_load_async_to_lds_b128(
      (gvec_ptr)(g + e + 4), (lvec_ptr)(l + e + 4), 0, 0);
#else
  #pragma unroll
  for (int t = 0; t < ELEMS_PER_THREAD; ++t) {
    const int i = tid + t * THREADS;
    l[i] = g[i];
  }
#endif
}

__device__ __forceinline__ void wait_async_done() {
#if __has_builtin(__builtin_amdgcn_global_load_async_to_lds_b128)
  #if __has_builtin(__builtin_amdgcn_s_wait_asynccnt)
    __builtin_amdgcn_s_wait_asynccnt(0);
  #else
    asm volatile("s_wait_asynccnt 0x0" ::: "memory");
  #endif
#endif
}

// Compare-exchange: ascending if up, else descending.
__device__ __forceinline__ void ce(float& a, float& b, bool up) {
  const float lo = fminf(a, b);
  const float hi = fmaxf(a, b);
  a = up ? lo : hi;
  b = up ? hi : lo;
}

// Bitonic tail j=4,2,1 over 8 registers, uniform direction.
__device__ __forceinline__ void tail421(float r[8], bool up) {
  ce(r[0], r[4], up); ce(r[1], r[5], up); ce(r[2], r[6], up); ce(r[3], r[7], up);
  ce(r[0], r[2], up); ce(r[1], r[3], up); ce(r[4], r[6], up); ce(r[5], r[7], up);
  ce(r[0], r[1], up); ce(r[2], r[3], up); ce(r[4], r[5], up); ce(r[6], r[7], up);
}

// Full bitonic sort of 8 registers; final direction d (asc if d).
// Directions follow global bitonic network for a chunk whose base is a
// multiple of 8: stages k=2,4 depend only on local index; k=8 uses d.
__device__ __forceinline__ void sort8(float r[8], bool d) {
  // k=2
  ce(r[0], r[1], true);  ce(r[2], r[3], false);
  ce(r[4], r[5], true);  ce(r[6], r[7], false);
  // k=4, j=2
  ce(r[0], r[2], true);  ce(r[1], r[3], true);
  ce(r[4], r[6], false); ce(r[5], r[7], false);
  // k=4, j=1
  ce(r[0], r[1], true);  ce(r[2], r[3], true);
  ce(r[4], r[5], false); ce(r[6], r[7], false);
  // k=8, j=4,2,1
  tail421(r, d);
}

__device__ __forceinline__ void load8(const float* s, int base, float r[8]) {
  const float4* p = reinterpret_cast<const float4*>(s + base);
  float4 a = p[0], b = p[1];
  r[0] = a.x; r[1] = a.y; r[2] = a.z; r[3] = a.w;
  r[4] = b.x; r[5] = b.y; r[6] = b.z; r[7] = b.w;
}

__device__ __forceinline__ void store8(float* s, int base, const float r[8]) {
  float4* p = reinterpret_cast<float4*>(s + base);
  p[0] = make_float4(r[0], r[1], r[2], r[3]);
  p[1] = make_float4(r[4], r[5], r[6], r[7]);
}

// One workgroup per row: async-load both rows into LDS, hybrid
// registers+LDS bitonic sort of both, L1 distance, per-row loss.
__global__ __launch_bounds__(THREADS)
void w1_rows_kernel(const float* __restrict__ pred,
                    const float* __restrict__ tru,
                    float* __restrict__ row_loss) {
  __shared__ float su[ROW_N];
  __shared__ float sv[ROW_N];

  const int tid  = threadIdx.x;
  const int row  = blockIdx.x;
  const int base = tid * ELEMS_PER_THREAD;
  const float* gu = pred + (size_t)row * ROW_N;
  const float* gv = tru  + (size_t)row * ROW_N;

  async_copy_row(gu, su, tid);
  async_copy_row(gv, sv, tid);
  wait_async_done();
  __syncthreads();

  float ru[8], rv[8];

  // ---- Phase A: stages k=2,4,8 fully in registers on the thread's chunk ----
  load8(su, base, ru);
  load8(sv, base, rv);
  const bool d0 = ((tid & 1) == 0);   // (base & 8) == 0
  sort8(ru, d0);
  sort8(rv, d0);
  store8(su, base, ru);
  store8(sv, base, rv);
  __syncthreads();

  // ---- Phase B: stages k=16..4096; j>=8 in LDS, j=4,2,1 in registers ----
  for (unsigned k = 16; k <= ROW_N; k <<= 1) {
    for (unsigned j = k >> 1; j >= 8; j >>= 1) {
      #pragma unroll
      for (int t = 0; t < PAIRS_PER_THREAD; ++t) {
        const unsigned p  = (unsigned)tid + (unsigned)t * THREADS; // pair id
        const unsigned i  = 2u * p - (p & (j - 1u));               // low index
        const unsigned m  = i + j;                                 // partner
        const bool     up = ((i & k) == 0u);

        float a = su[i], b = su[m];
        float lo = fminf(a, b), hi = fmaxf(a, b);
        su[i] = up ? lo : hi;
        su[m] = up ? hi : lo;

        float c = sv[i], d = sv[m];
        float lo2 = fminf(c, d), hi2 = fmaxf(c, d);
        sv[i] = up ? lo2 : hi2;
        sv[m] = up ? hi2 : lo2;
      }
      __syncthreads();
    }

    // Register tail: direction is uniform over the chunk for k >= 16.
    const bool up = (((unsigned)base & k) == 0u);
    load8(su, base, ru);
    load8(sv, base, rv);
    tail421(ru, up);
    tail421(rv, up);
    if (k < ROW_N) {            // final stage keeps data in registers
      store8(su, base, ru);
      store8(sv, base, rv);
      __syncthreads();
    }
  }

  // ---- L1 distance straight from registers (rows fully sorted) ----
  float acc = 0.0f;
  #pragma unroll
  for (int t = 0; t < 8; ++t) acc += fabsf(ru[t] - rv[t]);

  // Block tree reduction (reuse su; barrier so all chunk reads are done).
  __syncthreads();
  su[tid] = acc;
  __syncthreads();
  #pragma unroll
  for (int s = THREADS / 2; s > 0; s >>= 1) {
    if (tid < s) su[tid] += su[tid + s];
    __syncthreads();
  }
  if (tid == 0) row_loss[row] = su[0] * (1.0f / (float)ROW_N);
}

// Deterministic final mean over B row losses.
__global__ __launch_bounds__(256)
void w1_mean_kernel(const float* __restrict__ row_loss,
                    float* __restrict__ out, int B) {
  __shared__ float s[256];
  const int tid = threadIdx.x;
  float acc = 0.0f;
  for (int i = tid; i < B; i += 256) acc += row_loss[i];
  s[tid] = acc;
  __syncthreads();
  #pragma unroll
  for (int st = 128; st > 0; st >>= 1) {
    if (tid < st) s[tid] += s[tid + st];
    __syncthreads();
  }
  if (tid == 0) out[0] = s[0] / (float)B;
}

extern "C" void kernel_launch(void* const* d_in, const int* in_sizes, int n_in,
                              void* d_out, int out_size, void* d_ws, size_t ws_size,
                              hipStream_t stream) {
  const float* pred = (const float*)d_in[0];
  const float* tru  = (const float*)d_in[1];
  float* out        = (float*)d_out;
  float* row_loss   = (float*)d_ws;      // B floats of scratch

  const int B = in_sizes[0] / ROW_N;

  w1_rows_kernel<<<B, THREADS, 0, stream>>>(pred, tru, row_loss);
  w1_mean_kernel<<<1, 256, 0, stream>>>(row_loss, out, B);
}